// PlaquetteqGPS_73993696575820
// MI455X (gfx1250) — compile-verified
//
#include <hip/hip_runtime.h>
#include <hip/hip_bf16.h>
#include <math.h>

typedef __attribute__((ext_vector_type(2))) float v2f;
typedef __attribute__((ext_vector_type(8))) float v8f;

#define L_     64     // sites
#define S_     9      // plaquette width^2
#define NPLAQ  64
#define M_     128
#define BATCH_ 2048
#define KPAD   12     // 9 occ terms + 1 bias + 2 zero pad (3 K-steps of 4)
#define NTOT   256    // 128 magnitude cols || 128 parity cols

// ---------------------------------------------------------------------------
// Kernel 1: per-plaquette B matrices in log/sign domain.
//   Bw[p][k][n], k<9,n<128 : log|eps1|-log|eps0|   (magnitude diff)
//   Bw[p][k][128+n]        : signbit(e1)-signbit(e0)
//   k==9                   : base rows (sum of eps0 logs / sign count)
//   k==10,11               : zero padding to K=12
// epsilon layout: eps[o][p][m][s], o in {0,1}
// ---------------------------------------------------------------------------
__global__ void prep_kernel(const float* __restrict__ eps, float* __restrict__ Bw) {
    const int p = blockIdx.x;     // 0..63
    const int m = threadIdx.x;    // 0..127
    const float* e0 = eps + (size_t)((0 * NPLAQ + p) * M_ + m) * S_;
    const float* e1 = eps + (size_t)((1 * NPLAQ + p) * M_ + m) * S_;
    float* Bp = Bw + (size_t)p * KPAD * NTOT;

    float sumL = 0.f;
    int   sumS = 0;
#pragma unroll
    for (int s = 0; s < S_; ++s) {
        const float a = e0[s], b = e1[s];
        const float l0 = __logf(fabsf(a));
        const float l1 = __logf(fabsf(b));
        const int   s0 = (a < 0.f) ? 1 : 0;
        const int   s1 = (b < 0.f) ? 1 : 0;
        Bp[s * NTOT + m]       = l1 - l0;
        Bp[s * NTOT + 128 + m] = (float)(s1 - s0);
        sumL += l0;
        sumS += s0;
    }
    Bp[9 * NTOT + m]        = sumL;
    Bp[9 * NTOT + 128 + m]  = (float)sumS;
    Bp[10 * NTOT + m]       = 0.f;
    Bp[10 * NTOT + 128 + m] = 0.f;
    Bp[11 * NTOT + m]       = 0.f;
    Bp[11 * NTOT + 128 + m] = 0.f;
}

// ---------------------------------------------------------------------------
// Kernel 2: per (batch-tile of 16, plaquette) block.
//   A (16x12)  = [occ(b,p,s) | 1 | 0 0]  built from the spin gather
//   B (12x256) = prepped log/sign matrix
//   D = A@B via v_wmma_f32_16x16x4_f32, 3 K-steps per 16-col tile.
//   Epilogue: val = (-1)^parity * exp(mag); reduce over N; atomic to out[b].
// 4 waves; wave w owns magnitude cols [w*32, w*32+32) and paired parity cols.
// ---------------------------------------------------------------------------
__global__ void __launch_bounds__(128, 4)
gemm_kernel(const int* __restrict__ inputs, const int* __restrict__ plaq,
            const float* __restrict__ Bw, float* __restrict__ out) {
    const int p    = blockIdx.y;
    const int b0   = blockIdx.x * 16;
    const int tid  = threadIdx.x;
    const int wave = tid >> 5;
    const int lane = tid & 31;

    __shared__ float Alds[16][KPAD];
    __shared__ float acc[16];
    __shared__ int   plds[S_];

    if (tid < S_)  plds[tid] = plaq[p * S_ + tid];
    if (tid < 16)  acc[tid]  = 0.f;
    __syncthreads();

    // Build the shared A tile (16 rows x 12 K) from the gathered occupancies.
    for (int idx = tid; idx < 16 * KPAD; idx += 128) {
        const int r = idx / KPAD, k = idx % KPAD;
        float v;
        if (k < S_)       v = (float)inputs[(b0 + r) * L_ + plds[k]];
        else if (k == 9)  v = 1.f;   // bias row multiplies base[p,m]
        else              v = 0.f;   // K padding
        Alds[r][k] = v;
    }
    __syncthreads();

    // ISA 16x4 f32 A layout: lane L -> row = L&15; VGPR v -> K = 4*kk + 2*(L>>4) + v
    const int row  = lane & 15;
    const int half = lane >> 4;
    v2f a[3];
#pragma unroll
    for (int kk = 0; kk < 3; ++kk) {
        a[kk].x = Alds[row][kk * 4 + 2 * half + 0];
        a[kk].y = Alds[row][kk * 4 + 2 * half + 1];
    }

    const float* Bp = Bw + (size_t)p * KPAD * NTOT;

    v8f dmag[2], dpar[2];
#pragma unroll
    for (int t = 0; t < 2; ++t) {
        const int nm = wave * 32 + t * 16;   // magnitude tile cols
        const int np = 128 + nm;             // paired parity tile cols

        v8f cm = {}, cp = {};
#pragma unroll
        for (int kk = 0; kk < 3; ++kk) {
            // 4x16 f32 B layout mirrors A: lane L -> col = L&15; VGPR v -> K row 4*kk+2*(L>>4)+v
            const int kb = kk * 4 + 2 * half;
            v2f bm, bp;
            bm.x = Bp[(kb + 0) * NTOT + nm + row];
            bm.y = Bp[(kb + 1) * NTOT + nm + row];
            bp.x = Bp[(kb + 0) * NTOT + np + row];
            bp.y = Bp[(kb + 1) * NTOT + np + row];
            cm = __builtin_amdgcn_wmma_f32_16x16x4_f32(false, a[kk], false, bm,
                                                       (short)0, cm, false, false);
            cp = __builtin_amdgcn_wmma_f32_16x16x4_f32(false, a[kk], false, bp,
                                                       (short)0, cp, false, false);
        }
        dmag[t] = cm;
        dpar[t] = cp;
    }

    // Epilogue: sign*exp, reduce across the 16 N-lanes of each D row,
    // accumulate per-batch-row partials in LDS, then one atomic per row.
#pragma unroll
    for (int r = 0; r < 8; ++r) {
        float v = 0.f;
#pragma unroll
        for (int t = 0; t < 2; ++t) {
            const int cnt = (int)(dpar[t][r] + 0.5f);     // exact count in {0..9}
            const float sgn = (cnt & 1) ? -1.f : 1.f;
            v += sgn * __expf(dmag[t][r]);
        }
#pragma unroll
        for (int off = 1; off < 16; off <<= 1)
            v += __shfl_xor(v, off, 32);
        if ((lane & 15) == 0)                 // lane0 -> row r, lane16 -> row r+8
            atomicAdd(&acc[r + half * 8], v); // ds_add_f32
    }
    __syncthreads();
    if (tid < 16)
        atomicAdd(&out[b0 + tid], acc[tid]);  // global_atomic_add_f32, 64 p-blocks
}

// ---------------------------------------------------------------------------
extern "C" void kernel_launch(void* const* d_in, const int* in_sizes, int n_in,
                              void* d_out, int out_size, void* d_ws, size_t ws_size,
                              hipStream_t stream) {
    (void)in_sizes; (void)n_in; (void)out_size; (void)ws_size;
    const int*   inputs = (const int*)d_in[0];
    const int*   plaq   = (const int*)d_in[1];
    const float* eps    = (const float*)d_in[2];
    float* out = (float*)d_out;
    float* Bw  = (float*)d_ws;   // 64 * 12 * 256 * 4B = 768 KB

    hipMemsetAsync(d_out, 0, sizeof(float) * BATCH_, stream);
    prep_kernel<<<dim3(NPLAQ), dim3(M_), 0, stream>>>(eps, Bw);
    gemm_kernel<<<dim3(BATCH_ / 16, NPLAQ), dim3(128), 0, stream>>>(inputs, plaq, Bw, out);
}